// ScoreModel_83829171683978
// MI455X (gfx1250) — compile-verified
//
#include <hip/hip_runtime.h>
#include <math.h>

typedef __attribute__((ext_vector_type(2))) float v2f;
typedef __attribute__((ext_vector_type(8))) float v8f;

#define NSC 16
#define NVC 4
#define NNODES 50000
#define NEDGES 800000
#define EF 48
#define WN 368
#define OUTC 28           // NS + 3*NV
#define BN_EPS 1e-5f

// workspace layout (floats)
#define WS_SUMS 0
#define WS_CNT  (NNODES*OUTC)         // 1,400,000
#define WS_STAT (WS_CNT + NNODES)     // +50,000
#define WS_TOTAL (WS_STAT + 36)

// Relaxed device-scope FP32 add -> native GLOBAL_ATOMIC_ADD_F32 (no CAS loop,
// no return value -> non-returning atomic tracked on STOREcnt).
__device__ __forceinline__ void atomic_add_f32(float* p, float v) {
  (void)__hip_atomic_fetch_add(p, v, __ATOMIC_RELAXED, __HIP_MEMORY_SCOPE_AGENT);
}

__global__ void zero_ws(float* ws, int n) {
  int i = blockIdx.x * blockDim.x + threadIdx.x;
  if (i < n) ws[i] = 0.0f;
}

// ---------------------------------------------------------------------------
// Fused edge pass: GEMM1(relu) -> GEMM2 -> tensor-product tail -> atomic scatter
// One wave = 16 edges (M dim of WMMA). blockDim = 64 (2 waves).
// ---------------------------------------------------------------------------
__global__ __launch_bounds__(64) void edge_kernel(
    const float* __restrict__ node_attr,
    const int*   __restrict__ edge_index,   // [2][NEDGES]: src then dst
    const float* __restrict__ edge_attr,    // [NEDGES][48]
    const float* __restrict__ edge_sh,      // [NEDGES][9]
    const float* __restrict__ fc_w1,        // [48][48]
    const float* __restrict__ fc_b1,        // [48]
    const float* __restrict__ fc_w2,        // [48][368]
    const float* __restrict__ fc_b2,        // [368]
    float* __restrict__ sums,               // [NNODES][28] accum
    float* __restrict__ cnt)                // [NNODES]
{
  __shared__ float hBuf[2][16 * EF];   // per-wave h tile (16 x 48)
  __shared__ float wBuf[2][16 * WN];   // per-wave w tile (16 x 368)

  const int wv   = threadIdx.x >> 5;
  const int lane = threadIdx.x & 31;
  const int m    = lane & 15;    // matrix row (edge within tile) / col index
  const int half = lane >> 4;    // K-pair selector for A/B fragments
  const int e0   = (blockIdx.x * 2 + wv) * 16;

  // ---- A1 fragments: edge_attr rows, f32 WMMA 16x4 layout ----
  // lane(half,m): VGPR0 = A[m][4kk+2*half], VGPR1 = A[m][4kk+2*half+1]
  v2f a1[12];
  {
    const float* ea = edge_attr + (size_t)(e0 + m) * EF + 2 * half;
#pragma unroll
    for (int kk = 0; kk < 12; ++kk)
      a1[kk] = *(const v2f*)(ea + 4 * kk);
  }

  // ---- GEMM1: h = relu(A @ W1 + b1), N = 48 (3 tiles of 16) ----
#pragma unroll
  for (int t = 0; t < 3; ++t) {
    const float bias = fc_b1[16 * t + m];
    v8f c;
#pragma unroll
    for (int r = 0; r < 8; ++r) c[r] = bias;
#pragma unroll
    for (int kk = 0; kk < 12; ++kk) {
      const float* wp = fc_w1 + (size_t)(4 * kk + 2 * half) * EF + 16 * t + m;
      v2f b; b.x = wp[0]; b.y = wp[EF];
      c = __builtin_amdgcn_wmma_f32_16x16x4_f32(false, a1[kk], false, b,
                                                (short)0, c, false, false);
    }
    // D layout: VGPR r -> row (r + 8*half), col m
#pragma unroll
    for (int r = 0; r < 8; ++r)
      hBuf[wv][(r + 8 * half) * EF + 16 * t + m] = fmaxf(c[r], 0.0f);
  }
  __syncthreads();

  // ---- A2 fragments from LDS (same A for all 23 N-tiles) ----
  v2f a2[12];
#pragma unroll
  for (int kk = 0; kk < 12; ++kk)
    a2[kk] = *(const v2f*)(&hBuf[wv][m * EF + 4 * kk + 2 * half]);

  // ---- GEMM2: w = h @ W2 + b2, N = 368 (23 tiles of 16) ----
  for (int t2 = 0; t2 < 23; ++t2) {
    const float bias = fc_b2[16 * t2 + m];
    v8f c;
#pragma unroll
    for (int r = 0; r < 8; ++r) c[r] = bias;
#pragma unroll
    for (int kk = 0; kk < 12; ++kk) {
      const float* wp = fc_w2 + (size_t)(4 * kk + 2 * half) * WN + 16 * t2 + m;
      v2f b; b.x = wp[0]; b.y = wp[WN];
      c = __builtin_amdgcn_wmma_f32_16x16x4_f32(false, a2[kk], false, b,
                                                (short)0, c, false, false);
    }
#pragma unroll
    for (int r = 0; r < 8; ++r)
      wBuf[wv][(r + 8 * half) * WN + 16 * t2 + m] = c[r];
  }
  __syncthreads();

  // ---- per-edge tensor-product tail; 2 lanes per edge ----
  {
    const float* wr   = &wBuf[wv][m * WN];
    const int    e    = e0 + m;
    const int    srcn = edge_index[e];
    const int    dstn = edge_index[NEDGES + e];
    const float* na   = node_attr + (size_t)dstn * (NSC + 6);
    const float* shp  = edge_sh + (size_t)e * 9;
    const float  sh0  = shp[0];
    const float  sh1[3] = { shp[1], shp[2], shp[3] };

    float xs[16];
#pragma unroll
    for (int u = 0; u < 16; ++u) xs[u] = na[u];
    float xv[6];
#pragma unroll
    for (int i = 0; i < 6; ++i) xv[i] = na[16 + i];

    float* sacc = sums + (size_t)srcn * OUTC;
    const float inv_sqrt3  = 0.5773502691896258f;
    const float inv_sqrt18 = 0.23570226039551584f;
    const float inv_sqrt20 = 0.22360679774997896f;

    if (half == 0) {
      // scalar channel: p1 + p2
      const float dot0 = (xv[0]*sh1[0] + xv[1]*sh1[1] + xv[2]*sh1[2]) * inv_sqrt3;
      const float dot1 = (xv[3]*sh1[0] + xv[4]*sh1[1] + xv[5]*sh1[2]) * inv_sqrt3;
#pragma unroll
      for (int v = 0; v < 16; ++v) {
        float p1 = 0.0f;
#pragma unroll
        for (int u = 0; u < 16; ++u) p1 += xs[u] * wr[u * 16 + v];
        p1 *= sh0;
        const float p2 = dot0 * wr[328 + v] + dot1 * wr[344 + v];
        atomic_add_f32(&sacc[v], (p1 + p2) * inv_sqrt18);
      }
    } else {
      // vector channel: p3 + p4 + p5
      float t3[4] = {0.f, 0.f, 0.f, 0.f};
#pragma unroll
      for (int u = 0; u < 16; ++u) {
        const float x = xs[u];
#pragma unroll
        for (int v = 0; v < 4; ++v) t3[v] += x * wr[256 + u * 4 + v];
      }
      // M[i][j] = sum_m sh2[m] * W3J121[m][i][j]
      const float s20 = shp[4], s21 = shp[5], s22 = shp[6], s23 = shp[7], s24 = shp[8];
      const float c10  = 0.31622776601683794f;  // 1/sqrt(10)
      const float c30  = 0.18257418583505536f;  // 1/sqrt(30)
      const float c30b = 0.36514837167011072f;  // 2/sqrt(30)
      const float M00 = -c30 * s22 + c10 * s24;
      const float M01 =  c10 * s20;
      const float M02 =  c10 * s23;
      const float M10 =  c10 * s20;
      const float M11 = -c30 * s22 - c10 * s24;
      const float M12 =  c10 * s21;
      const float M20 =  c10 * s23;
      const float M21 =  c10 * s21;
      const float M22 =  c30b * s22;

      float vec[2][3];
#pragma unroll
      for (int u = 0; u < 2; ++u) {
        vec[u][0] = xv[u*3+0]*M00 + xv[u*3+1]*M10 + xv[u*3+2]*M20;
        vec[u][1] = xv[u*3+0]*M01 + xv[u*3+1]*M11 + xv[u*3+2]*M21;
        vec[u][2] = xv[u*3+0]*M02 + xv[u*3+1]*M12 + xv[u*3+2]*M22;
      }
#pragma unroll
      for (int v = 0; v < 4; ++v) {
        const float w4a = wr[320 + v], w4b = wr[324 + v];
        const float w5a = wr[360 + v], w5b = wr[364 + v];
#pragma unroll
        for (int i = 0; i < 3; ++i) {
          const float p3 = t3[v] * sh1[i];
          const float p4 = sh0 * (xv[i] * w4a + xv[3 + i] * w4b);
          const float p5 = vec[0][i] * w5a + vec[1][i] * w5b;
          atomic_add_f32(&sacc[NSC + v * 3 + i], (p3 + p4 + p5) * inv_sqrt20);
        }
      }
      atomic_add_f32(&cnt[srcn], 1.0f);
    }
  }
}

// ---------------------------------------------------------------------------
// Per-node mean + batch-norm statistics (wave-shuffle reduce -> LDS -> global)
// ---------------------------------------------------------------------------
__global__ __launch_bounds__(256) void node_stats(
    float* __restrict__ sums,          // in: raw sums, out: per-node mean
    const float* __restrict__ cnt,
    float* __restrict__ stat)          // [36]: sum_s[16], sumsq_s[16], sum_vsq[4]
{
  __shared__ float st[36];
  const int tid = threadIdx.x;
  if (tid < 36) st[tid] = 0.0f;
  __syncthreads();

  const int  n     = blockIdx.x * blockDim.x + tid;
  const bool valid = (n < NNODES);
  float inv = 1.0f;
  if (valid) inv = 1.0f / fmaxf(cnt[n], 1.0f);

  float vals[OUTC];
#pragma unroll
  for (int j = 0; j < OUTC; ++j) {
    float v = 0.0f;
    if (valid) {
      v = sums[(size_t)n * OUTC + j] * inv;
      sums[(size_t)n * OUTC + j] = v;
    }
    vals[j] = v;
  }

  const int lane = tid & 31;
#pragma unroll
  for (int c = 0; c < NSC; ++c) {
    float s = vals[c], s2 = vals[c] * vals[c];
    for (int off = 16; off > 0; off >>= 1) {
      s  += __shfl_down(s,  off, 32);
      s2 += __shfl_down(s2, off, 32);
    }
    if (lane == 0) { atomicAdd(&st[c], s); atomicAdd(&st[16 + c], s2); }
  }
#pragma unroll
  for (int u = 0; u < NVC; ++u) {
    float vs = 0.0f;
#pragma unroll
    for (int i = 0; i < 3; ++i) { const float x = vals[NSC + u * 3 + i]; vs += x * x; }
    for (int off = 16; off > 0; off >>= 1) vs += __shfl_down(vs, off, 32);
    if (lane == 0) atomicAdd(&st[32 + u], vs);
  }
  __syncthreads();
  if (tid < 36) atomic_add_f32(&stat[tid], st[tid]);
}

// ---------------------------------------------------------------------------
// Apply batch norm, write final output
// ---------------------------------------------------------------------------
__global__ __launch_bounds__(256) void node_finalize(
    const float* __restrict__ avg,     // [NNODES][28]
    const float* __restrict__ stat,    // [36]
    const float* __restrict__ bn_weight,  // [20]
    const float* __restrict__ bn_bias,    // [16]
    float* __restrict__ out)           // [NNODES][28]
{
  const int n = blockIdx.x * blockDim.x + threadIdx.x;
  if (n >= NNODES) return;
  const float invN = 1.0f / (float)NNODES;
#pragma unroll
  for (int c = 0; c < NSC; ++c) {
    const float mean = stat[c] * invN;
    const float var  = stat[16 + c] * invN - mean * mean;
    const float sc   = rsqrtf(var + BN_EPS) * bn_weight[c];
    const float v    = avg[(size_t)n * OUTC + c];
    out[(size_t)n * OUTC + c] = (v - mean) * sc + bn_bias[c];
  }
#pragma unroll
  for (int u = 0; u < NVC; ++u) {
    const float vn = stat[32 + u] * invN * (1.0f / 3.0f);
    const float sc = rsqrtf(vn + BN_EPS) * bn_weight[NSC + u];
#pragma unroll
    for (int i = 0; i < 3; ++i) {
      const size_t idx = (size_t)n * OUTC + NSC + u * 3 + i;
      out[idx] = avg[idx] * sc;
    }
  }
}

extern "C" void kernel_launch(void* const* d_in, const int* in_sizes, int n_in,
                              void* d_out, int out_size, void* d_ws, size_t ws_size,
                              hipStream_t stream) {
  (void)in_sizes; (void)n_in; (void)out_size; (void)ws_size;
  const float* node_attr  = (const float*)d_in[0];
  const int*   edge_index = (const int*)  d_in[1];
  const float* edge_attr  = (const float*)d_in[2];
  const float* edge_sh    = (const float*)d_in[3];
  const float* fc_w1      = (const float*)d_in[4];
  const float* fc_b1      = (const float*)d_in[5];
  const float* fc_w2      = (const float*)d_in[6];
  const float* fc_b2      = (const float*)d_in[7];
  const float* bn_weight  = (const float*)d_in[8];
  const float* bn_bias    = (const float*)d_in[9];

  float* ws   = (float*)d_ws;
  float* sums = ws + WS_SUMS;
  float* cntb = ws + WS_CNT;
  float* stat = ws + WS_STAT;

  zero_ws<<<(WS_TOTAL + 255) / 256, 256, 0, stream>>>(ws, WS_TOTAL);
  edge_kernel<<<NEDGES / 32, 64, 0, stream>>>(node_attr, edge_index, edge_attr,
                                              edge_sh, fc_w1, fc_b1, fc_w2, fc_b2,
                                              sums, cntb);
  node_stats<<<(NNODES + 255) / 256, 256, 0, stream>>>(sums, cntb, stat);
  node_finalize<<<(NNODES + 255) / 256, 256, 0, stream>>>(sums, stat, bn_weight,
                                                          bn_bias, (float*)d_out);
}